// Decoder_41858751267150
// MI455X (gfx1250) — compile-verified
//
#include <hip/hip_runtime.h>

// D = A(16x4 f32) * B(4x16 f32) + C(16x16 f32) via V_WMMA_F32_16X16X4_F32
typedef __attribute__((ext_vector_type(2))) float v2f;
typedef __attribute__((ext_vector_type(8))) float v8f;

#define B_ 4
#define N_ 512
#define K_ 16000
#define L_ 16
#define STEP_ 8                      // L/2
#define T_ (STEP_ * (K_ - 1) + L_)   // 128008
#define KTILE_ 16
#define TILES_PER_B_ (K_ / KTILE_)   // 1000
#define WAVES_TOTAL_ (B_ * TILES_PER_B_)  // 4000
#define BLOCK_ 256                   // 8 waves (wave32)

__global__ __launch_bounds__(BLOCK_) void zero_f32(float* __restrict__ p, int n) {
    int i = blockIdx.x * BLOCK_ + threadIdx.x;
    if (i < n) p[i] = 0.0f;
}

// One wave computes frames tile [16 l  x 16 k] for one b, then overlap-adds
// with global f32 atomics: frames[b,k,l] -> out[b, 8k + l].
__global__ __launch_bounds__(BLOCK_) void decoder_wmma_f32(
    const float* __restrict__ mix,   // [B, N, K]
    const float* __restrict__ mask,  // [B, N, K]
    const float* __restrict__ W,     // [L, N]
    float* __restrict__ out)         // [B, T]
{
    const int wave = (int)((blockIdx.x * (unsigned)BLOCK_ + threadIdx.x) >> 5);
    if (wave >= WAVES_TOTAL_) return;            // wave-uniform: EXEC stays all-1s
    const int lane = (int)(threadIdx.x & 31u);

    const int b  = wave / TILES_PER_B_;
    const int k0 = (wave % TILES_PER_B_) * KTILE_;

    const int col = lane & 15;   // A: row M=l ; B: column (k within tile)
    const int hi  = lane >> 4;   // 0 => reduction slots K=0,1 ; 1 => K=2,3

    // A-matrix source: W[l = col][n0 + 2*hi + {0,1}]  (N contiguous -> float2)
    const float* wrow = W + (size_t)col * N_ + 2 * hi;
    // B-matrix source: est[n0 + 2*hi + {0,1}][k0 + col], est = mix * mask.
    // For a fixed n, the 16 lower (or upper) lanes read 16 consecutive k -> 64B segments.
    const size_t base = ((size_t)b * N_ + 2 * hi) * K_ + (size_t)(k0 + col);
    const float* mixp  = mix  + base;
    const float* maskp = mask + base;

    v8f c = {};
    #pragma unroll 4
    for (int n0 = 0; n0 < N_; n0 += 4) {
        v2f a;
        a.x = wrow[n0];
        a.y = wrow[n0 + 1];

        const size_t o0 = (size_t)n0 * K_;
        const size_t o1 = o0 + K_;
        float m0 = __builtin_nontemporal_load(mixp  + o0);
        float m1 = __builtin_nontemporal_load(mixp  + o1);
        float s0 = __builtin_nontemporal_load(maskp + o0);
        float s1 = __builtin_nontemporal_load(maskp + o1);

        v2f bv;
        bv.x = m0 * s0;
        bv.y = m1 * s1;

        // (neg_a, A, neg_b, B, c_mod, C, reuse_a, reuse_b)
        c = __builtin_amdgcn_wmma_f32_16x16x4_f32(
                false, a, false, bv, (short)0, c, false, false);
    }

    // D layout: VGPR j -> M = j + 8*hi, column = col. frames[k0+col][M] -> t = 8*(k0+col)+M
    float* outb = out + (size_t)b * T_;
    const int tbase = STEP_ * (k0 + col) + 8 * hi;
    #pragma unroll
    for (int j = 0; j < 8; ++j) {
        atomicAdd(outb + tbase + j, c[j]);
    }
}

extern "C" void kernel_launch(void* const* d_in, const int* in_sizes, int n_in,
                              void* d_out, int out_size, void* d_ws, size_t ws_size,
                              hipStream_t stream) {
    const float* mix  = (const float*)d_in[0];  // mixture_w [B,N,K] f32
    const float* mask = (const float*)d_in[1];  // est_mask  [B,N,K] f32
    const float* W    = (const float*)d_in[2];  // W [L,N] f32
    float* out        = (float*)d_out;          // [B,T] f32

    const int n_out = B_ * T_;
    zero_f32<<<(n_out + BLOCK_ - 1) / BLOCK_, BLOCK_, 0, stream>>>(out, n_out);

    const int waves_per_block = BLOCK_ / 32;                       // 8
    const int grid = WAVES_TOTAL_ / waves_per_block;               // 500 exact
    decoder_wmma_f32<<<grid, BLOCK_, 0, stream>>>(mix, mask, W, out);
}